// LocalFeatureBuilder_15668040696417
// MI455X (gfx1250) — compile-verified
//
#include <hip/hip_runtime.h>
#include <math.h>

typedef __attribute__((ext_vector_type(2))) float v2f;
typedef __attribute__((ext_vector_type(8))) float v8f;

#define N_ATOMS   32768
#define N_QUERIES 8192
#define K_NEIGH   32
#define EMBED     64
#define FEAT      133
#define CHUNKS    (N_ATOMS / 16)   // 2048 WMMA column chunks
#define ROWSTRIDE 20               // dwords; 80B rows -> 16B aligned for b128

// ---------------------------------------------------------------------------
// Kernel 0: repack atom coords into WMMA-B lane layout + per-column |c|^2.
// B operand for V_WMMA_F32_16X16X4_F32 is 4x16 f32 = 2 VGPRs:
//   VGPR0: lanes 0-15 -> K=0 row (x of atoms 0..15), lanes 16-31 -> K=2 (z)
//   VGPR1: lanes 0-15 -> K=1 row (y),                lanes 16-31 -> K=3 (0)
// Stored as float2 per (chunk, lane) so the hot loop does one b64 load.
// ---------------------------------------------------------------------------
__global__ void prep_kernel(const float* __restrict__ coords,
                            float* __restrict__ wsB,   // CHUNKS*32 float2
                            float* __restrict__ wsC2)  // CHUNKS*32 float
{
    int t = blockIdx.x * blockDim.x + threadIdx.x;
    if (t >= CHUNKS * 32) return;
    int lane = t & 31;
    int col  = lane & 15;
    int chunk = t >> 5;
    int a = chunk * 16 + col;
    float x = coords[3 * a + 0];
    float y = coords[3 * a + 1];
    float z = coords[3 * a + 2];
    float v0 = (lane < 16) ? x : z;
    float v1 = (lane < 16) ? y : 0.0f;
    wsB[2 * t + 0] = v0;
    wsB[2 * t + 1] = v1;
    wsC2[t] = x * x + y * y + z * z;
}

// ---------------------------------------------------------------------------
// Kernel 1: fused WMMA distance GEMM + register-resident top-32 per query.
// One wave (32 lanes) owns 32 queries = two 16-row WMMA tiles.
//   d^2 = |q|^2 (in C operand) + (-2q).c (WMMA) + |c|^2 (VALU add)
// Tiles are transposed through LDS (row stride 20 dw: 16B-aligned rows,
// conflict-free stores, 2-way-max loads), candidates pulled with 4x b128,
// and a 16-wide min tree gates the sorted-insertion network so the
// steady-state chunk cost is WMMA + LDS only.
// ---------------------------------------------------------------------------
__global__ void __launch_bounds__(32)
topk_kernel(const float* __restrict__ query,
            const float* __restrict__ wsB,
            const float* __restrict__ wsC2,
            float* __restrict__ out_mask,   // [Q*K]
            int*   __restrict__ out_idx,    // [Q*K]
            float* __restrict__ out_dist)   // [Q*K]
{
    __shared__ float lds[32 * ROWSTRIDE];

    const int lane  = threadIdx.x;
    const int col   = lane & 15;
    const int hi    = lane >> 4;          // 0: lanes 0-15, 1: lanes 16-31
    const int qbase = blockIdx.x * 32;

    // A operands (16x4, -2*q folded in), one per 16-query tile.
    v2f a0, a1;
    {
        int q0 = qbase + col;
        int q1 = qbase + 16 + col;
        float x0 = query[3 * q0 + 0], y0 = query[3 * q0 + 1], z0 = query[3 * q0 + 2];
        float x1 = query[3 * q1 + 0], y1 = query[3 * q1 + 1], z1 = query[3 * q1 + 2];
        a0.x = hi ? (-2.0f * z0) : (-2.0f * x0);   // K=0 / K=2
        a0.y = hi ? 0.0f         : (-2.0f * y0);   // K=1 / K=3 (pad)
        a1.x = hi ? (-2.0f * z1) : (-2.0f * x1);
        a1.y = hi ? 0.0f         : (-2.0f * y1);
    }

    // C operands pre-loaded with |q|^2 per destination row.
    v8f c0, c1;
#pragma unroll
    for (int i = 0; i < 8; ++i) {
        int r0 = qbase + i + 8 * hi;
        int r1 = qbase + 16 + i + 8 * hi;
        float x = query[3 * r0 + 0], y = query[3 * r0 + 1], z = query[3 * r0 + 2];
        c0[i] = x * x + y * y + z * z;
        x = query[3 * r1 + 0]; y = query[3 * r1 + 1]; z = query[3 * r1 + 2];
        c1[i] = x * x + y * y + z * z;
    }

    // Register-resident sorted top-32 (d^2 ascending).
    float top_d[K_NEIGH];
    int   top_i[K_NEIGH];
#pragma unroll
    for (int j = 0; j < K_NEIGH; ++j) { top_d[j] = 3.0e38f; top_i[j] = 0; }

    const v2f*    __restrict__ bptr = (const v2f*)wsB;
    const float4* __restrict__ rowp = (const float4*)&lds[lane * ROWSTRIDE];

    float worst = 3.0e38f;

    for (int chunk = 0; chunk < CHUNKS; ++chunk) {
        v2f   b  = bptr[chunk * 32 + lane];
        float c2 = wsC2[chunk * 32 + lane];
        // Speculative prefetch of next chunk's B stream (OOB is dropped).
        __builtin_prefetch(&bptr[(chunk + 1) * 32 + lane], 0, 1);

        v8f d0 = __builtin_amdgcn_wmma_f32_16x16x4_f32(
            false, a0, false, b, (short)0, c0, false, false);
        v8f d1 = __builtin_amdgcn_wmma_f32_16x16x4_f32(
            false, a1, false, b, (short)0, c1, false, false);

        // d^2 tiles -> LDS (row = query-in-wave).
#pragma unroll
        for (int i = 0; i < 8; ++i) {
            lds[(i + 8 * hi) * ROWSTRIDE + col]      = d0[i] + c2;
            lds[(16 + i + 8 * hi) * ROWSTRIDE + col] = d1[i] + c2;
        }
        __syncthreads();

        // Pull this lane's 16 candidates with 4x ds_load_b128, one wait.
        float4 g0 = rowp[0];
        float4 g1 = rowp[1];
        float4 g2 = rowp[2];
        float4 g3 = rowp[3];

        float cand[16] = { g0.x, g0.y, g0.z, g0.w,
                           g1.x, g1.y, g1.z, g1.w,
                           g2.x, g2.y, g2.z, g2.w,
                           g3.x, g3.y, g3.z, g3.w };

        // 16-wide min tree gates the insertion network (rare after warm-up).
        float mn = fminf(fminf(fminf(cand[0], cand[1]),  fminf(cand[2],  cand[3])),
                         fminf(fminf(cand[4], cand[5]),  fminf(cand[6],  cand[7])));
        mn = fminf(mn,
             fminf(fminf(fminf(cand[8],  cand[9]),  fminf(cand[10], cand[11])),
                   fminf(fminf(cand[12], cand[13]), fminf(cand[14], cand[15]))));

        if (mn < worst) {
#pragma unroll 4
            for (int j = 0; j < 16; ++j) {
                float d2 = cand[j];
                if (d2 < worst) {
                    float cd = d2;
                    int   ci = chunk * 16 + j;
#pragma unroll
                    for (int s = 0; s < K_NEIGH; ++s) {
                        bool  sw = cd < top_d[s];
                        float od = top_d[s];
                        int   oi = top_i[s];
                        top_d[s] = sw ? cd : od;
                        top_i[s] = sw ? ci : oi;
                        cd = sw ? od : cd;
                        ci = sw ? oi : ci;
                    }
                    worst = top_d[K_NEIGH - 1];
                }
            }
        }
        __syncthreads();
    }

    // Emit sorted dists / indices / mask for this lane's query.
    int q = qbase + lane;
#pragma unroll 4
    for (int j = 0; j < K_NEIGH; ++j) {
        float dist = sqrtf(fmaxf(top_d[j], 1e-12f));
        float m = (dist <= 8.0f) ? 1.0f : 0.0f;
        if (j == 0) m = 1.0f;
        out_dist[q * K_NEIGH + j] = dist;
        out_idx [q * K_NEIGH + j] = top_i[j];
        out_mask[q * K_NEIGH + j] = m;
    }
}

// ---------------------------------------------------------------------------
// Kernel 2: streaming feature build. One wave per (query, neighbor) pair,
// coalesced 133-float row writes.  ~139 MB -> HBM-bandwidth bound (~6 us).
// RBF: gamma = 1/(8/64)^2 = 64, centers = 8*r/63.
// ---------------------------------------------------------------------------
__global__ void feat_kernel(const float* __restrict__ coords,
                            const int*   __restrict__ atom_types,
                            const float* __restrict__ radii,
                            const float* __restrict__ query,
                            const float* __restrict__ emb,
                            const float* __restrict__ mask_r,
                            const int*   __restrict__ idx_r,
                            const float* __restrict__ dist_r,
                            float* __restrict__ feat)
{
    int gtid = blockIdx.x * blockDim.x + threadIdx.x;
    int pair = gtid >> 5;
    int lane = threadIdx.x & 31;
    if (pair >= N_QUERIES * K_NEIGH) return;

    int   q    = pair >> 5;  // / K_NEIGH
    int   idx  = idx_r[pair];
    float dist = dist_r[pair];
    float m    = mask_r[pair];

    float rx = query[3 * q + 0] - coords[3 * idx + 0];
    float ry = query[3 * q + 1] - coords[3 * idx + 1];
    float rz = query[3 * q + 2] - coords[3 * idx + 2];
    float rad = radii[idx];
    int   at  = atom_types[idx];

    float* __restrict__ frow = feat + (size_t)pair * FEAT;
#pragma unroll
    for (int t = 0; t < 5; ++t) {
        int f = lane + 32 * t;
        if (f >= FEAT) break;
        float v;
        if (f < 3) {
            v = (f == 0) ? rx : ((f == 1) ? ry : rz);
        } else if (f == 3) {
            v = rad;
        } else if (f < 4 + EMBED) {
            v = emb[at * EMBED + (f - 4)];
        } else if (f < 132) {
            float c = 8.0f * (float)(f - 68) * (1.0f / 63.0f);
            float e = dist - c;
            v = __expf(-64.0f * e * e);
        } else {
            v = dist;
        }
        frow[f] = v * m;
    }
}

// ---------------------------------------------------------------------------
extern "C" void kernel_launch(void* const* d_in, const int* in_sizes, int n_in,
                              void* d_out, int out_size, void* d_ws, size_t ws_size,
                              hipStream_t stream)
{
    const float* coords     = (const float*)d_in[0];
    const int*   atom_types = (const int*)  d_in[1];
    const float* radii      = (const float*)d_in[2];
    const float* query      = (const float*)d_in[3];
    const float* emb        = (const float*)d_in[4];

    // d_out layout: features | mask | sorted_indices | sorted_dists
    float* out   = (float*)d_out;
    float* featp = out;
    float* maskp = featp + (size_t)N_QUERIES * K_NEIGH * FEAT;
    int*   idxp  = (int*)(maskp + (size_t)N_QUERIES * K_NEIGH);
    float* distp = (float*)(idxp + (size_t)N_QUERIES * K_NEIGH);

    // Workspace: B-layout coords (CHUNKS*32 float2) + |c|^2 (CHUNKS*32 floats)
    float* wsB  = (float*)d_ws;
    float* wsC2 = wsB + (size_t)CHUNKS * 32 * 2;

    prep_kernel<<<(CHUNKS * 32 + 255) / 256, 256, 0, stream>>>(coords, wsB, wsC2);

    topk_kernel<<<N_QUERIES / 32, 32, 0, stream>>>(query, wsB, wsC2,
                                                   maskp, idxp, distp);

    int pairs = N_QUERIES * K_NEIGH;
    feat_kernel<<<(pairs * 32) / 256, 256, 0, stream>>>(coords, atom_types, radii,
                                                        query, emb,
                                                        maskp, idxp, distp, featp);
}